// SpatialPositionalEncoding_29953101922423
// MI455X (gfx1250) — compile-verified
//
#include <hip/hip_runtime.h>
#include <hip/hip_bf16.h>

// MI455X / gfx1250, wave32. Fused pairwise-MLP + masked mean + projection.
// Heavy math (N^2 x 64 x 64 layer) runs on V_WMMA_F32_16X16X4_F32 chains.
// W2 B-fragments live in 128 VGPRs per lane (no LDS/VMEM in the hot loop).
// Best-measured configuration (compile histogram): s-outer A-frag build,
// post-relu vj multiply, no register cap. Both the vj-into-features fold and
// a forced 256-VGPR cap were tried and produced worse schedules (spills /
// +SALU), so this exact structure is intentional.

#define NOBJ 2048
#define DHID 64
#define DMODEL 256

typedef float v2f __attribute__((ext_vector_type(2)));
typedef float v8f __attribute__((ext_vector_type(8)));

// ---------------- Phase 0: per-object geometry ----------------
__global__ void spe_geom_kernel(const float* __restrict__ corners,
                                const unsigned char* __restrict__ vmask,
                                float4* __restrict__ geo,
                                float* __restrict__ validf) {
  int i = blockIdx.x * blockDim.x + threadIdx.x;
  if (i >= NOBJ) return;
  const float* c = corners + i * 24;
  float sx = 0.f, sy = 0.f, sz = 0.f;
  float mnx = 1e30f, mny = 1e30f, mnz = 1e30f;
  float mxx = -1e30f, mxy = -1e30f, mxz = -1e30f;
#pragma unroll
  for (int k = 0; k < 8; ++k) {
    float x = c[k * 3 + 0], y = c[k * 3 + 1], z = c[k * 3 + 2];
    sx += x; sy += y; sz += z;
    mnx = fminf(mnx, x); mny = fminf(mny, y); mnz = fminf(mnz, z);
    mxx = fmaxf(mxx, x); mxy = fmaxf(mxy, y); mxz = fmaxf(mxz, z);
  }
  float ex = fmaxf(mxx - mnx, 1e-6f);
  float ey = fmaxf(mxy - mny, 1e-6f);
  float ez = fmaxf(mxz - mnz, 1e-6f);
  float lv = logf(ex * ey * ez + 1e-6f);
  geo[i] = make_float4(sx * 0.125f, sy * 0.125f, sz * 0.125f, lv);
  validf[i] = (vmask[i] != 0) ? 1.0f : 0.0f;
}

// ---------------- Phase 1: fused pairwise MLP + masked sum ----------------
// Block = 256 threads = 8 waves; each wave owns one row i, loops 128 j-tiles.
__global__ __launch_bounds__(256) void spe_pair_kernel(
    const float4* __restrict__ geo, const float* __restrict__ validf,
    const float* __restrict__ W1, const float* __restrict__ b1,
    const float* __restrict__ W2, const float* __restrict__ b2,
    float* __restrict__ agg) {
  // Packed W1/b1: per k-pair group g (k=2g,2g+1): 12 floats
  // {W1[0..4][2g], b1[2g], W1[0..4][2g+1], b1[2g+1]}
  alignas(16) __shared__ float w1p[32 * 12];
  int tid = threadIdx.x;
  if (tid < DHID) {
    int k = tid, g = k >> 1, slot = (k & 1) * 6;
#pragma unroll
    for (int q = 0; q < 5; ++q) w1p[g * 12 + slot + q] = W1[q * DHID + k];
    w1p[g * 12 + slot + 5] = b1[k];
  }
  __syncthreads();

  int wave = tid >> 5, lane = tid & 31;
  int i = blockIdx.x * 8 + wave;
  int p = lane & 15, hi = lane >> 4;

  float4 gi = geo[i];
  float vi = validf[i];

  // per-lane output-column bias (col = t*16 + p)
  float b2v[4];
#pragma unroll
  for (int t = 0; t < 4; ++t) b2v[t] = b2[t * 16 + p];

  // Preload W2 B-fragments: bf[t][s] = {W2[4s+2hi][t*16+p], W2[4s+2hi+1][t*16+p]}
  v2f bf[4][16];
#pragma unroll
  for (int t = 0; t < 4; ++t) {
    int col = t * 16 + p;
#pragma unroll
    for (int s = 0; s < 16; ++s) {
      int k0 = 4 * s + 2 * hi;
      v2f b;
      b.x = W2[k0 * DHID + col];
      b.y = W2[(k0 + 1) * DHID + col];
      bf[t][s] = b;
    }
  }

  float acc[4] = {0.f, 0.f, 0.f, 0.f};
  float cntv = 0.f;

  for (int jt = 0; jt < NOBJ / 16; ++jt) {
    int j = jt * 16 + p;               // both half-waves mirror p
    float4 gj = geo[j];
    float vj = validf[j];
    cntv += vj;

    float dx = gi.x - gj.x, dy = gi.y - gj.y, dz = gi.z - gj.z;
    float dist = sqrtf(dx * dx + dy * dy + dz * dz);
    float inv = 1.0f / (dist + 1e-6f);
    float f0 = dist, f1 = dx * inv, f2 = dy * inv, f3 = dz * inv;
    float f4 = gi.w - gj.w;

    v8f d0 = {}, d1 = {}, d2 = {}, d3 = {};

    // s-outer: build one A-fragment, feed 4 independent WMMA chains, drop it.
#pragma unroll
    for (int s = 0; s < 16; ++s) {
      int g = 2 * s + hi;  // k0 = 2g = 4s+2hi
      const float4* wp = (const float4*)&w1p[g * 12];
      float4 A = wp[0], B = wp[1], C = wp[2];
      float z0 = B.y + f0 * A.x + f1 * A.y + f2 * A.z + f3 * A.w + f4 * B.x;
      float z1 = C.w + f0 * B.z + f1 * B.w + f2 * C.x + f3 * C.y + f4 * C.z;
      v2f a;
      a.x = fmaxf(z0, 0.f) * vj;
      a.y = fmaxf(z1, 0.f) * vj;
      d0 = __builtin_amdgcn_wmma_f32_16x16x4_f32(false, a, false, bf[0][s], (short)0, d0, false, false);
      d1 = __builtin_amdgcn_wmma_f32_16x16x4_f32(false, a, false, bf[1][s], (short)0, d1, false, false);
      d2 = __builtin_amdgcn_wmma_f32_16x16x4_f32(false, a, false, bf[2][s], (short)0, d2, false, false);
      d3 = __builtin_amdgcn_wmma_f32_16x16x4_f32(false, a, false, bf[3][s], (short)0, d3, false, false);
    }

    // bias + relu + row-sum (hi=0 -> rows 0..7, hi=1 -> rows 8..15)
    float s0 = 0.f, s1 = 0.f, s2 = 0.f, s3 = 0.f;
#pragma unroll
    for (int r = 0; r < 8; ++r) {
      s0 += fmaxf(d0[r] + b2v[0], 0.f);
      s1 += fmaxf(d1[r] + b2v[1], 0.f);
      s2 += fmaxf(d2[r] + b2v[2], 0.f);
      s3 += fmaxf(d3[r] + b2v[3], 0.f);
    }
    acc[0] += s0; acc[1] += s1; acc[2] += s2; acc[3] += s3;
  }

  // cnt = sum_j validf[j] (reduce over the 16 lanes of each half)
  float c = cntv;
  c += __shfl_xor(c, 1, 32);
  c += __shfl_xor(c, 2, 32);
  c += __shfl_xor(c, 4, 32);
  c += __shfl_xor(c, 8, 32);
  float nval = fmaxf(c, 1.0f);
  float ninv = (float)NOBJ - c;  // invalid-j rows each contributed relu(b2)

#pragma unroll
  for (int t = 0; t < 4; ++t) {
    float tot = acc[t] + __shfl_xor(acc[t], 16, 32);
    float rb2 = fmaxf(b2v[t], 0.f);
    float outv = (vi > 0.f) ? (tot - ninv * rb2) / nval : 0.f;
    if (hi == 0) agg[i * DHID + t * 16 + p] = outv;
  }
}

// ---------------- Phase 2: out = agg @ Wo + bo (WMMA f32) ----------------
__global__ __launch_bounds__(256) void spe_proj_kernel(
    const float* __restrict__ agg, const float* __restrict__ Wo,
    const float* __restrict__ bo, float* __restrict__ out) {
  int wave = threadIdx.x >> 5, lane = threadIdx.x & 31;
  int wid = blockIdx.x * 8 + wave;       // 0..2047 tiles
  int mt = wid >> 4, nt = wid & 15;      // 128 M-tiles x 16 N-tiles
  int p = lane & 15, hi = lane >> 4;
  int col = nt * 16 + p;

  v8f d = {};
#pragma unroll
  for (int s = 0; s < 16; ++s) {
    int k0 = 4 * s + 2 * hi;
    const float* ap = agg + (mt * 16 + p) * DHID + k0;
    v2f a; a.x = ap[0]; a.y = ap[1];
    v2f b;
    b.x = Wo[k0 * DMODEL + col];
    b.y = Wo[(k0 + 1) * DMODEL + col];
    d = __builtin_amdgcn_wmma_f32_16x16x4_f32(
        false, a, false, b, (short)0, d, false, false);
  }
  float bc = bo[col];
#pragma unroll
  for (int r = 0; r < 8; ++r)
    out[(mt * 16 + r + 8 * hi) * DMODEL + col] = d[r] + bc;
}

// ---------------- launcher ----------------
extern "C" void kernel_launch(void* const* d_in, const int* in_sizes, int n_in,
                              void* d_out, int out_size, void* d_ws, size_t ws_size,
                              hipStream_t stream) {
  (void)in_sizes; (void)n_in; (void)out_size; (void)ws_size;
  const float* corners        = (const float*)d_in[0];          // (2048,8,3) f32
  const unsigned char* vmask  = (const unsigned char*)d_in[1];  // (2048,) bool
  const float* W1 = (const float*)d_in[2];   // (5,64)
  const float* b1 = (const float*)d_in[3];   // (64,)
  const float* W2 = (const float*)d_in[4];   // (64,64)
  const float* b2 = (const float*)d_in[5];   // (64,)
  const float* Wo = (const float*)d_in[6];   // (64,256)
  const float* bo = (const float*)d_in[7];   // (256,)
  float* out = (float*)d_out;                // (2048,256)

  // Workspace layout: geo (32KB) | validf (8KB) | agg (512KB)
  char* ws = (char*)d_ws;
  float4* geo   = (float4*)ws;
  float* validf = (float*)(ws + 32768);
  float* agg    = (float*)(ws + 40960);

  spe_geom_kernel<<<NOBJ / 256, 256, 0, stream>>>(corners, vmask, geo, validf);
  spe_pair_kernel<<<NOBJ / 8, 256, 0, stream>>>(geo, validf, W1, b1, W2, b2, agg);
  spe_proj_kernel<<<(128 * 16) / 8, 256, 0, stream>>>(agg, Wo, bo, out);
}